// StarGAT_metapath_specific_14035953123580
// MI455X (gfx1250) — compile-verified
//
#include <hip/hip_runtime.h>
#include <hip/hip_bf16.h>

typedef __attribute__((ext_vector_type(2))) float v2f;
typedef __attribute__((ext_vector_type(8))) float v8f;

#define DIM   64
#define LPATH 4
#define ALPHA 0.01f

// ---- monotonic float<->int key for atomicMax on signed int ----
__device__ __forceinline__ int f32_key(float f) {
  int i = __float_as_int(f);
  return i >= 0 ? i : (i ^ 0x7FFFFFFF);
}
__device__ __forceinline__ float key_f32(int k) {
  return __int_as_float(k >= 0 ? k : (k ^ 0x7FFFFFFF));
}

// ---- pass 0: init workspace ----
__global__ void init_ws(int* __restrict__ a_max_key, float* __restrict__ a_sum,
                        float* __restrict__ ft, int N) {
  int t = blockIdx.x * blockDim.x + threadIdx.x;
  if (t < N) { a_max_key[t] = (int)0x80000000; a_sum[t] = 0.0f; }
  if (t < N * DIM) ft[t] = 0.0f;
}

// ---- pass 1: per-edge logit via WMMA f32 16x16x4 chain (K=64) ----
// One wave handles a tile of 16 edges. Lane pair (l, l+16) owns edge l%16:
// lane low holds eft elements {4t, 4t+1}, lane high {4t+2, 4t+3} -> exactly
// the A-matrix 16x4 VGPR layout for iteration kbase=4t.
__global__ void __launch_bounds__(256)
edge_logits_wmma(const float* __restrict__ feat, const float* __restrict__ attn,
                 const int* __restrict__ edge, const int* __restrict__ edge_dst,
                 float* __restrict__ a_out, int* __restrict__ a_max_key, int E) {
  const int lane = threadIdx.x & 31;
  const long wave = (long)((blockIdx.x * blockDim.x + threadIdx.x) >> 5);
  const long e0 = wave * 16;
  if (e0 >= E) return;                 // wave-uniform exit (EXEC stays full)
  const int m    = lane & 15;
  const int hi   = lane >> 4;          // 0: K%4 in {0,1}; 1: K%4 in {2,3}
  const int koff = hi * 2;

  long e = e0 + m;
  if (e >= E) e = (long)E - 1;         // clamp: keep all lanes active for WMMA

  const int* ep = edge + e * LPATH;
  const float* r0 = feat + (size_t)ep[0] * DIM;
  const float* r1 = feat + (size_t)ep[1] * DIM;
  const float* r2 = feat + (size_t)ep[2] * DIM;
  const float* r3 = feat + (size_t)ep[3] * DIM;

  v2f acc[16];
#pragma unroll
  for (int t = 0; t < 16; ++t) {
    const int o = 4 * t + koff;
    v2f v0 = *(const v2f*)(r0 + o);
    v2f v1 = *(const v2f*)(r1 + o);
    v2f v2 = *(const v2f*)(r2 + o);
    v2f v3 = *(const v2f*)(r3 + o);
    acc[t] = (v0 + v1 + v2 + v3) * 0.25f;   // mean over 4 path positions
  }

  // Chain 16 WMMAs: D = A(16x4 eft chunk) x B(4x16 attn bcast) + C.
  // B[k][n] = attn[4t+k] for all n -> every output column equals dot(eft,attn).
  v8f c = {0.f, 0.f, 0.f, 0.f, 0.f, 0.f, 0.f, 0.f};
#pragma unroll
  for (int t = 0; t < 16; ++t) {
    v2f b = *(const v2f*)(attn + 4 * t + koff);
    c = __builtin_amdgcn_wmma_f32_16x16x4_f32(false, acc[t], false, b,
                                              (short)0, c, false, false);
  }

  // C/D layout: VGPR r, lanes 0-15 -> M=r; lanes 16-31 -> M=r+8 (any column).
  if (m == 0) {
#pragma unroll
    for (int r = 0; r < 8; ++r) {
      long ee = e0 + r + 8 * hi;
      if (ee < E) {
        float av = c[r];
        av = av > 0.0f ? av : ALPHA * av;   // LeakyReLU
        a_out[ee] = av;
        atomicMax(&a_max_key[edge_dst[ee]], f32_key(av));
      }
    }
  }
}

// ---- pass 2: exp(a - max) and per-dst sum ----
__global__ void softmax_expsum(const int* __restrict__ edge_dst,
                               const int* __restrict__ a_max_key,
                               float* __restrict__ a, float* __restrict__ a_sum,
                               int E) {
  long e = (long)blockIdx.x * blockDim.x + threadIdx.x;
  if (e >= E) return;
  int dst = edge_dst[e];
  float ex = __expf(a[e] - key_f32(a_max_key[dst]));
  a[e] = ex;
  atomicAdd(&a_sum[dst], ex);
}

// ---- pass 3: recompute eft from L2-resident features, weighted scatter ----
__global__ void __launch_bounds__(256)
weighted_scatter(const float* __restrict__ feat, const int* __restrict__ edge,
                 const int* __restrict__ edge_dst, const float* __restrict__ a_exp,
                 const float* __restrict__ a_sum, float* __restrict__ ft, int E) {
  const int lane = threadIdx.x & 31;
  const long wave = (long)((blockIdx.x * blockDim.x + threadIdx.x) >> 5);
  const long e0 = wave * 16;
  if (e0 >= E) return;
  const int m    = lane & 15;
  const int hi   = lane >> 4;
  const int koff = hi * 2;

  long e = e0 + m;
  const bool valid = (e < E);
  if (!valid) e = (long)E - 1;

  const int* ep = edge + e * LPATH;
  const int dst = edge_dst[e];
  // fold mean (0.25) and softmax weight into one scale; 0 for clamped tail lanes
  const float w = valid ? (a_exp[e] / a_sum[dst]) * 0.25f : 0.0f;

  const float* r0 = feat + (size_t)ep[0] * DIM;
  const float* r1 = feat + (size_t)ep[1] * DIM;
  const float* r2 = feat + (size_t)ep[2] * DIM;
  const float* r3 = feat + (size_t)ep[3] * DIM;
  float* base = ft + (size_t)dst * DIM;

#pragma unroll
  for (int t = 0; t < 16; ++t) {
    const int o = 4 * t + koff;
    v2f v0 = *(const v2f*)(r0 + o);
    v2f v1 = *(const v2f*)(r1 + o);
    v2f v2 = *(const v2f*)(r2 + o);
    v2f v3 = *(const v2f*)(r3 + o);
    v2f v  = (v0 + v1 + v2 + v3) * w;
    atomicAdd(base + o,     v.x);
    atomicAdd(base + o + 1, v.y);
  }
}

// ---- pass 4: gather target rows ----
__global__ void gather_out(const float* __restrict__ ft, const int* __restrict__ tix,
                           float* __restrict__ out, long total) {
  long t = (long)blockIdx.x * blockDim.x + threadIdx.x;
  if (t >= total) return;
  long row = t >> 6;
  int  col = (int)(t & 63);
  out[t] = ft[(size_t)tix[row] * DIM + col];
}

extern "C" void kernel_launch(void* const* d_in, const int* in_sizes, int n_in,
                              void* d_out, int out_size, void* d_ws, size_t ws_size,
                              hipStream_t stream) {
  const float* feat     = (const float*)d_in[0];
  const float* attn     = (const float*)d_in[1];
  const int*   edge     = (const int*)d_in[2];
  const int*   edge_dst = (const int*)d_in[3];
  const int*   tix      = (const int*)d_in[4];

  const int N = in_sizes[0] / DIM;   // 100000
  const int E = in_sizes[3];         // 1600000
  const int T = in_sizes[4];         // 50000

  // workspace layout (floats): a[E] | a_max_key[N] | a_sum[N] | ft[N*DIM]
  float* ws        = (float*)d_ws;
  float* a         = ws;
  int*   a_max_key = (int*)(ws + (size_t)E);
  float* a_sum     = ws + (size_t)E + N;
  float* ft        = ws + (size_t)E + 2 * (size_t)N;
  float* out       = (float*)d_out;

  {
    int total = N * DIM;
    init_ws<<<(total + 255) / 256, 256, 0, stream>>>(a_max_key, a_sum, ft, N);
  }
  {
    int tiles  = (E + 15) / 16;
    int blocks = (tiles + 7) / 8;    // 8 waves (tiles) per 256-thread block
    edge_logits_wmma<<<blocks, 256, 0, stream>>>(feat, attn, edge, edge_dst,
                                                 a, a_max_key, E);
  }
  softmax_expsum<<<(E + 255) / 256, 256, 0, stream>>>(edge_dst, a_max_key, a,
                                                      a_sum, E);
  {
    int tiles  = (E + 15) / 16;
    int blocks = (tiles + 7) / 8;
    weighted_scatter<<<blocks, 256, 0, stream>>>(feat, edge, edge_dst, a,
                                                 a_sum, ft, E);
  }
  {
    long total = (long)T * DIM;
    gather_out<<<(int)((total + 255) / 256), 256, 0, stream>>>(ft, tix, out, total);
  }
}